// _DynamicCudaWrapper_2542620639816
// MI455X (gfx1250) — compile-verified
//
#include <hip/hip_runtime.h>

// CSR SpMV for MI455X (gfx1250, wave32).
// out[r] = sum_{j in [start_r, off[r+1])} vals[j] * x[col[j]],
// start_0 = 0, start_r = off[r] for r >= 1 (matches jnp.searchsorted side='right'
// over off[1:]; off[0] is ignored, nnz >= off[num_rows] are dropped).
//
// Bandwidth-bound: ~264 MB of single-use stream traffic vs a 4 MB gather target
// that lives in the 192 MB L2. Strategy:
//   * wave-per-row CSR-vector, coalesced b32 streams of vals/col_idx
//   * non-temporal loads on the streams (keep x resident in L2/WGP$)
//   * global_prefetch_b8 to warm each row's stream
//   * block-level async global->LDS staging of the row_end slice (ASYNCcnt path)
//   * wave32 butterfly reduction with __shfl_xor, NT store of the result

#define WAVES_PER_BLOCK 8
#define BLOCK_SIZE (WAVES_PER_BLOCK * 32)

#if __has_builtin(__builtin_amdgcn_global_load_async_to_lds_b32) && \
    __has_builtin(__builtin_amdgcn_s_wait_asynccnt)
#define USE_ASYNC_LDS 1
#else
#define USE_ASYNC_LDS 0
#endif

typedef __attribute__((address_space(1))) int* gint_ptr;
typedef __attribute__((address_space(3))) int* lint_ptr;

__global__ __launch_bounds__(BLOCK_SIZE) void spmv_csr_vector_w32(
    const float* __restrict__ vals,
    const float* __restrict__ x,
    const int*   __restrict__ col,
    const int*   __restrict__ row_end,   // NUM_ROWS + 1 entries
    float*       __restrict__ out,
    int num_rows)
{
    __shared__ int s_off[WAVES_PER_BLOCK + 1];

    const int lane = threadIdx.x & 31;
    const int wave = threadIdx.x >> 5;
    const int row0 = blockIdx.x * WAVES_PER_BLOCK;

    // Stage this block's 9 row offsets into LDS once, instead of 2 SMEM loads
    // per wave. Async copy is EXEC-masked: only lanes 0..8 issue transfers.
    if (threadIdx.x <= WAVES_PER_BLOCK) {
        int r = row0 + threadIdx.x;
        if (r > num_rows) r = num_rows;   // clamp: row_end has num_rows+1 entries
#if USE_ASYNC_LDS
        // global (AS1) and LDS (AS3) operand pointers; low 32 bits of a generic
        // LDS pointer are the LDS byte offset (ISA aperture rule).
        __builtin_amdgcn_global_load_async_to_lds_b32(
            (gint_ptr)(unsigned long long)(row_end + r),
            (lint_ptr)(unsigned int)(unsigned long long)(&s_off[threadIdx.x]),
            0, 0);
#else
        s_off[threadIdx.x] = row_end[r];
#endif
    }
#if USE_ASYNC_LDS
    __builtin_amdgcn_s_wait_asynccnt(0);   // issuing wave drains its ASYNCcnt
#endif
    __syncthreads();                        // LDS now visible to all 8 waves

    const int row = row0 + wave;            // wave-uniform
    if (row >= num_rows) return;

    const int start = (row == 0) ? 0 : s_off[wave];
    const int end   = s_off[wave + 1];

    // Warm the streaming lines for this row (speculative prefetch into GL2).
    if (lane == 0 && start < end) {
        __builtin_prefetch(vals + start, 0, 1);
        __builtin_prefetch(col  + start, 0, 1);
    }

    float acc = 0.0f;
    for (int j = start + lane; j < end; j += 32) {
        // vals/col are read exactly once across the whole dispatch:
        // non-temporal so the 256 MB stream does not evict the 4 MB x from L2.
        const float v = __builtin_nontemporal_load(vals + j);
        const int   c = __builtin_nontemporal_load(col  + j);
        acc = fmaf(v, x[c], acc);   // x: regular temporal load -> L2/WGP$ hit
    }

    // wave32 butterfly reduction
    #pragma unroll
    for (int off = 16; off > 0; off >>= 1)
        acc += __shfl_xor(acc, off, 32);

    if (lane == 0)
        __builtin_nontemporal_store(acc, out + row);
}

extern "C" void kernel_launch(void* const* d_in, const int* in_sizes, int n_in,
                              void* d_out, int out_size, void* d_ws, size_t ws_size,
                              hipStream_t stream)
{
    const float* vals    = (const float*)d_in[0];   // NNZ f32
    const float* x       = (const float*)d_in[1];   // NUM_COLS f32
    const int*   col     = (const int*)  d_in[2];   // NNZ i32
    const int*   row_end = (const int*)  d_in[3];   // NUM_ROWS+1 i32

    float* out = (float*)d_out;
    const int num_rows = in_sizes[3] - 1;

    const int grid = (num_rows + WAVES_PER_BLOCK - 1) / WAVES_PER_BLOCK;
    spmv_csr_vector_w32<<<grid, BLOCK_SIZE, 0, stream>>>(
        vals, x, col, row_end, out, num_rows);
}